// MoEAggregator_455266533835
// MI455X (gfx1250) — compile-verified
//
#include <hip/hip_runtime.h>

typedef __attribute__((ext_vector_type(2))) float v2f;
typedef __attribute__((ext_vector_type(4))) float v4f;
typedef __attribute__((ext_vector_type(8))) float v8f;

static constexpr int B_    = 2;
static constexpr int S_    = 2048;
static constexpr int D_    = 4096;
static constexpr int N_    = 8;
static constexpr int TOPK_ = 2;

// ---------------------------------------------------------------------------
// Kernel 1: gating GEMV (2x8 = pooled(2,4096) x gate_W(8,4096)^T) via
// V_WMMA_F32_16X16X4_F32, + top-2 selection. One block, 8 waves.
//   A (16x4 f32): row m = batch (m<2 valid), lane = {m, khi}, VGPR0=K(2*khi),
//                 VGPR1=K(2*khi+1)           (ISA 7.12.2, 32-bit A 16x4)
//   B (4x16 f32): col n = adapter (n<8 valid), same lane striping mirrored.
//   C/D (16x16 f32): VGPR r holds row M=r (lanes 0-15) / M=r+8 (lanes 16-31);
//                 we only need acc[0] (M=0) and acc[1] (M=1).
// Invalid lanes (m>=B / n>=N) load from a clamped (valid) row and are zeroed
// with selects -> no EXEC manipulation in the loop, loads pipeline freely,
// and WMMA always executes with EXEC all-1s.
// ---------------------------------------------------------------------------
__global__ __launch_bounds__(256) void moe_gate_topk_wmma(
    const float* __restrict__ x,        // (B,S,D)
    const float* __restrict__ gate_W,   // (N,D)
    const float* __restrict__ gate_b,   // (N)
    int* __restrict__ idx_out)          // (B,TOPK) in workspace
{
  const int lane = threadIdx.x & 31;
  const int wave = threadIdx.x >> 5;
  const int m    = lane & 15;   // A row (batch) / B col (adapter)
  const int khi  = lane >> 4;   // K-pair select within a 4-wide K step

  const bool va = (m < B_);
  const bool vb = (m < N_);
  // Clamped, loop-invariant per-lane base pointers (always legal addresses).
  const float* aPtr = x + ((size_t)(va ? m : 0) * S_ + (S_ - 1)) * D_ + 2 * khi;
  const float* bPtr = gate_W + (size_t)(vb ? m : 0) * D_ + 2 * khi;

  // Two independent accumulator chains per wave to cover the WMMA->WMMA
  // RAW hazard / latency (compiler also inserts required NOPs).
  v8f acc0 = {};
  v8f acc1 = {};
  const v2f zero = {};

  const int steps = D_ / 4;     // 1024 K-steps of 4
  for (int j = wave * 2; j < steps; j += 16) {
    {
      v2f a = *(const v2f*)(aPtr + j * 4);
      v2f b = *(const v2f*)(bPtr + j * 4);
      a = va ? a : zero;
      b = vb ? b : zero;
      acc0 = __builtin_amdgcn_wmma_f32_16x16x4_f32(
          false, a, false, b, (short)0, acc0, false, false);
    }
    if (j + 1 < steps) {
      v2f a = *(const v2f*)(aPtr + (j + 1) * 4);
      v2f b = *(const v2f*)(bPtr + (j + 1) * 4);
      a = va ? a : zero;
      b = vb ? b : zero;
      acc1 = __builtin_amdgcn_wmma_f32_16x16x4_f32(
          false, a, false, b, (short)0, acc1, false, false);
    }
  }

  // Cross-wave reduction of the (2 x 16) valid accumulator slice.
  __shared__ float red[8][2][16];
  if (lane < 16) {
    red[wave][0][lane] = acc0[0] + acc1[0];  // batch 0, adapter=lane
    red[wave][1][lane] = acc0[1] + acc1[1];  // batch 1, adapter=lane
  }
  __syncthreads();

  if (threadIdx.x == 0) {
    for (int b = 0; b < B_; ++b) {
      float g[N_];
      for (int n = 0; n < N_; ++n) {
        float s = gate_b[n];
        for (int w = 0; w < 8; ++w) s += red[w][b][n];
        g[n] = s;
      }
      int best = 0;
      for (int n = 1; n < N_; ++n)
        if (g[n] > g[best]) best = n;
      int second = (best == 0) ? 1 : 0;
      for (int n = 0; n < N_; ++n) {
        if (n == best) continue;
        if (g[n] > g[second]) second = n;
      }
      idx_out[b * TOPK_ + 0] = best;
      idx_out[b * TOPK_ + 1] = second;
    }
  }
}

// ---------------------------------------------------------------------------
// Kernel 2: out = base_res + lora[..., i0] + lora[..., i1]
// Pure HBM-bound gather/add: b128 base loads, stride-8-float b32 lora
// gathers (layout-forced), nontemporal b128 output stores (touch-once data).
// 4 elements per thread. ~670 MB total traffic -> ~29 us floor @ 23.3 TB/s.
// ---------------------------------------------------------------------------
__global__ __launch_bounds__(256) void moe_gather_add(
    const float* __restrict__ base,   // (B,S,D)
    const float* __restrict__ lora,   // (B,S,D,N)
    const int*   __restrict__ idx,    // (B,TOPK)
    float* __restrict__ out)          // (B,S,D)
{
  const size_t e = ((size_t)blockIdx.x * blockDim.x + threadIdx.x) * 4;
  const int b = (int)(e / ((size_t)S_ * D_));   // uniform per block (SD % 1024 == 0)

  int i0 = idx[b * TOPK_ + 0];
  int i1 = idx[b * TOPK_ + 1];
  i0 = __builtin_amdgcn_readfirstlane(i0);      // wave-uniform -> SGPR
  i1 = __builtin_amdgcn_readfirstlane(i1);

  const v4f bv = *(const v4f*)(base + e);
  const float* lp = lora + e * N_;

  v4f r;
  r.x = bv.x + lp[0 * N_ + i0] + lp[0 * N_ + i1];
  r.y = bv.y + lp[1 * N_ + i0] + lp[1 * N_ + i1];
  r.z = bv.z + lp[2 * N_ + i0] + lp[2 * N_ + i1];
  r.w = bv.w + lp[3 * N_ + i0] + lp[3 * N_ + i1];

  __builtin_nontemporal_store(r, (v4f*)(out + e));
}

// ---------------------------------------------------------------------------
extern "C" void kernel_launch(void* const* d_in, const int* in_sizes, int n_in,
                              void* d_out, int out_size, void* d_ws, size_t ws_size,
                              hipStream_t stream) {
  const float* x        = (const float*)d_in[0];  // (B,S,D)
  const float* base_res = (const float*)d_in[1];  // (B,S,D)
  const float* lora     = (const float*)d_in[2];  // (B,S,D,N)
  const float* gate_W   = (const float*)d_in[3];  // (N,D)
  const float* gate_b   = (const float*)d_in[4];  // (N)
  // d_in[5] = top_k scalar (fixed at 2 by the reference)

  int* idx = (int*)d_ws;  // (B, TOPK) selected adapters

  moe_gate_topk_wmma<<<dim3(1), dim3(256), 0, stream>>>(x, gate_W, gate_b, idx);

  const size_t total  = (size_t)B_ * S_ * D_;       // 16,777,216
  const int    blocks = (int)(total / (4 * 256));   // 16,384
  moe_gather_add<<<dim3(blocks), dim3(256), 0, stream>>>(base_res, lora, idx,
                                                         (float*)d_out);
}